// SumPooling_84078279786855
// MI455X (gfx1250) — compile-verified
//
#include <hip/hip_runtime.h>

typedef __attribute__((ext_vector_type(2))) float v2f;
typedef __attribute__((ext_vector_type(4))) float v4f;
typedef __attribute__((ext_vector_type(8))) float v8f;
typedef __attribute__((ext_vector_type(4))) unsigned int v4u;
typedef __attribute__((ext_vector_type(4))) int v4i;
typedef __attribute__((ext_vector_type(8))) int v8i;

static constexpr int D = 128;
static constexpr int LDS_STRIDE = 132;  // 128 + 4 pad floats -> conflict-free ds_load_b64

// Detect which TDM builtin arity this toolchain has: the therock-10.0 headers
// (6-arg builtin) ship amd_gfx1250_TDM.h; ROCm 7.2 (5-arg builtin) does not.
#if !__has_include(<hip/amd_detail/amd_gfx1250_TDM.h>)
#define USE_TDM_5ARG 1
#else
#define USE_TDM_5ARG 0
#endif

// ---------------------------------------------------------------------------
// Kernel 1: zero the accumulator (d_out doubles as the pooled buffer).
// ---------------------------------------------------------------------------
__global__ void zero_f4(float* __restrict__ out, int n4) {
  int i = blockIdx.x * blockDim.x + threadIdx.x;
  if (i < n4) {
    reinterpret_cast<v4f*>(out)[i] = v4f{0.f, 0.f, 0.f, 0.f};
  }
}

// ---------------------------------------------------------------------------
// Kernel 2: segment scatter-add. One wave per input row (32 lanes x float4
// = 128 floats = one row). x/idx are streamed once -> non-temporal loads so
// they don't evict the L2-resident 32MB accumulator. f32 atomics resolve in
// the 192MB L2 (pooled is only 32MB).  HBM roofline: ~2.15GB / 23.3TB/s.
// ---------------------------------------------------------------------------
__global__ void scatter_add(const float* __restrict__ x,
                            const int* __restrict__ idx,
                            float* __restrict__ pooled,
                            unsigned nrows) {
  unsigned gid = blockIdx.x * blockDim.x + threadIdx.x;
  unsigned row = gid >> 5;
  if (row >= nrows) return;
  unsigned lane = gid & 31u;

  v4f v = __builtin_nontemporal_load(reinterpret_cast<const v4f*>(x) + gid);
  int seg = __builtin_nontemporal_load(idx + row);

  float* dst = pooled + (size_t)seg * D + lane * 4;
  unsafeAtomicAdd(dst + 0, v.x);
  unsafeAtomicAdd(dst + 1, v.y);
  unsafeAtomicAdd(dst + 2, v.z);
  unsafeAtomicAdd(dst + 3, v.w);
}

// ---------------------------------------------------------------------------
// Kernel 3: in-place fp32 WMMA GEMM.  out[i,j] = sum_k pooled[i,k]*W[j,k] + b[j]
// W (64KB) is staged in LDS once per block via the Tensor Data Mover with a
// padded row stride of 132 floats (TDM pad: interval=128 dwords, amount=4),
// making the B-operand ds_load_b64 pattern bank-conflict-free.
// One wave owns a 16x128 output stripe: 8 f32 accumulators, 32 k-steps,
// 256 x V_WMMA_F32_16X16X4_F32.
// ---------------------------------------------------------------------------
__global__ void gemm_wmma_inplace(float* __restrict__ io,       // [S,128] pooled in, result out
                                  const float* __restrict__ Wg, // [128,128] row-major [out,in]
                                  const float* __restrict__ bias,
                                  int srows) {
  __shared__ float sW[D * LDS_STRIDE];  // 67,584 bytes

#if USE_TDM_5ARG
  if ((threadIdx.x >> 5) == 0) {
    // ---- Tensor DMA descriptor (D#), ISA 8.3/8.4 ----
    const unsigned lds_base = (unsigned)(size_t)(&sW[0]);
    const unsigned long long ga = (unsigned long long)(size_t)Wg;
    v4u g0;
    g0.x = 1u;                                   // count=1, user descriptor
    g0.y = lds_base;                             // lds_addr
    g0.z = (unsigned)ga;                         // global_addr[31:0]
    g0.w = (unsigned)(ga >> 32) | (2u << 30);    // global_addr[56:32] | type=2
    v8i g1;
    g1[0] = (int)((2u << 16)      // data_size = 4 bytes
                | (1u << 20)      // pad_enable
                | (6u << 22)      // pad_interval: 2^(6+1) = 128 dwords
                | (3u << 25));    // pad_amount : 3+1 = 4 dwords  -> stride 132
    g1[1] = (int)(128u << 16);    // tensor_dim0 = 128 (low 16 bits)
    g1[2] = (int)(128u << 16);    // tensor_dim1 = 128 (low 16 bits)
    g1[3] = (int)(128u << 16);    // tile_dim0   = 128
    g1[4] = 128;                  // tile_dim1   = 128, tile_dim2 = 0
    g1[5] = 128;                  // tensor_dim0_stride = 128 elements
    g1[6] = 0;
    g1[7] = 0;
    v4i z4 = {0, 0, 0, 0};        // groups 2/3: tile_dim3/4 = 0 (unused)
    __builtin_amdgcn_tensor_load_to_lds(g0, g1, z4, z4, 0);
    __builtin_amdgcn_s_wait_tensorcnt(0);
  }
#else
  // Cooperative fallback: coalesced float4 copy into the padded LDS image.
  for (int v = threadIdx.x; v < D * (D / 4); v += blockDim.x) {
    int row = v >> 5;
    int c4  = v & 31;
    v4f t = *reinterpret_cast<const v4f*>(Wg + row * D + c4 * 4);
    *reinterpret_cast<v4f*>(&sW[row * LDS_STRIDE + c4 * 4]) = t;
  }
#endif
  __syncthreads();

  const int lane = threadIdx.x & 31;
  const int wave = threadIdx.x >> 5;
  const int row0 = (blockIdx.x * (blockDim.x >> 5) + wave) * 16;
  if (row0 >= srows) return;

  const int mlane = lane & 15;
  const int koff  = (lane >> 4) << 1;  // 0 for lanes 0-15, 2 for lanes 16-31

  v8f acc[8];
#pragma unroll
  for (int n = 0; n < 8; ++n) acc[n] = v8f{0.f, 0.f, 0.f, 0.f, 0.f, 0.f, 0.f, 0.f};

  const float* arow = io + (size_t)(row0 + mlane) * D;

  // Software-pipelined A load: next step's global load overlaps 8 WMMAs.
  v2f a_next = *reinterpret_cast<const v2f*>(arow + koff);

  for (int k0 = 0; k0 < D; k0 += 4) {
    v2f a = a_next;
    if (k0 + 4 < D) a_next = *reinterpret_cast<const v2f*>(arow + k0 + 4 + koff);

    // Batch all 8 B tiles (conflict-free ds_load_b64) before the WMMA burst.
    v2f bm[8];
#pragma unroll
    for (int n = 0; n < 8; ++n) {
      bm[n] = *reinterpret_cast<const v2f*>(
          &sW[(n * 16 + mlane) * LDS_STRIDE + k0 + koff]);
    }
#pragma unroll
    for (int n = 0; n < 8; ++n) {
      acc[n] = __builtin_amdgcn_wmma_f32_16x16x4_f32(
          /*neg_a=*/false, a, /*neg_b=*/false, bm[n],
          /*c_mod=*/(short)0, acc[n], /*reuse_a=*/false, /*reuse_b=*/false);
    }
  }

  const int rbase = row0 + ((lane >> 4) << 3);  // +0 or +8
#pragma unroll
  for (int n = 0; n < 8; ++n) {
    const float bv = bias[n * 16 + mlane];
#pragma unroll
    for (int r = 0; r < 8; ++r) {
      io[(size_t)(rbase + r) * D + n * 16 + mlane] = acc[n][r] + bv;
    }
  }
}

// ---------------------------------------------------------------------------
extern "C" void kernel_launch(void* const* d_in, const int* in_sizes, int n_in,
                              void* d_out, int out_size, void* d_ws, size_t ws_size,
                              hipStream_t stream) {
  const float* x    = (const float*)d_in[0];
  const int*   idx  = (const int*)d_in[1];
  // d_in[2] is dst_size (device scalar); S derived from out_size instead.
  const float* W    = (const float*)d_in[3];
  const float* bias = (const float*)d_in[4];
  float*       out  = (float*)d_out;

  const int N = in_sizes[0] / D;   // 4,194,304 rows
  const int S = out_size / D;      // 65,536 segments

  // 1) zero accumulator
  const int n4 = out_size / 4;
  zero_f4<<<(n4 + 255) / 256, 256, 0, stream>>>(out, n4);

  // 2) scatter-add: one wave per row, 4 floats per lane
  const unsigned total = (unsigned)N * 32u;
  scatter_add<<<(total + 255) / 256, 256, 0, stream>>>(x, idx, out, (unsigned)N);

  // 3) in-place fp32 WMMA transform: 8 waves/block, each wave = 16 rows
  const int waves_per_block = 8;
  const int row_blocks = S / 16;
  gemm_wmma_inplace<<<(row_blocks + waves_per_block - 1) / waves_per_block,
                      waves_per_block * 32, 0, stream>>>(out, W, bias, S);
}